// FusedLocalNormalizedCrossCorrelationLoss_62285615727506
// MI455X (gfx1250) — compile-verified
//
#include <hip/hip_runtime.h>

// ---------------------------------------------------------------------------
// Fused local normalized cross-correlation loss, 5x5x5 box window,
// input (2,1,192,192,192) fp32 -> scalar.
// Single fused pass:
//   - double-buffered async global->LDS halo-plane loads (ASYNCcnt pipeline)
//   - fused field compute (I,J,I2,J2,IJ) + x box-sum in VALU
//   - y box-sum as banded GEMM on the matrix pipe (V_WMMA_F32_16X16X4_F32)
//   - z box-sum via 5-slot LDS ring + pointwise NCC
//   - block reduce + global_atomic_add_f32 scalar
// ---------------------------------------------------------------------------

#ifndef __has_builtin
#define __has_builtin(x) 0
#endif

#if __has_builtin(__builtin_amdgcn_global_load_async_to_lds_b32)
#define HAVE_ASYNC_LDS 1
#else
#define HAVE_ASYNC_LDS 0
#endif

#if __has_builtin(__builtin_amdgcn_wmma_f32_16x16x4_f32)
#define HAVE_WMMA_F32 1
#else
#define HAVE_WMMA_F32 0
#endif

// Compile-time probes (device pass only; visible in stderr on success).
#if defined(__HIP_DEVICE_COMPILE__)
#if HAVE_ASYNC_LDS
#warning "CDNA5 probe: global_load_async_to_lds builtin ACTIVE"
#else
#warning "CDNA5 probe: global_load_async_to_lds builtin MISSING (sync fallback)"
#endif
#if HAVE_WMMA_F32
#warning "CDNA5 probe: wmma_f32_16x16x4_f32 builtin ACTIVE"
#else
#warning "CDNA5 probe: wmma_f32_16x16x4_f32 builtin MISSING (VALU fallback)"
#endif
#endif

typedef __attribute__((ext_vector_type(2))) float v2f;
typedef __attribute__((ext_vector_type(8))) float v8f;

#define KS       5
#define TILE     16
#define HALO     2
#define IN       20            // TILE + 2*HALO
#define ZCHUNK   24            // 192 / 24 = 8 chunks per batch
#define NIT      (ZCHUNK + 2 * HALO)
#define NTHREADS 256
#define DIM      192

#if HAVE_ASYNC_LDS
typedef __attribute__((address_space(1))) int g_i32;
typedef __attribute__((address_space(3))) int l_i32;
__device__ __forceinline__ void async_cp_g2l(const float* g, float* l) {
    __builtin_amdgcn_global_load_async_to_lds_b32(
        (g_i32*)g, (l_i32*)l, /*imm offset*/0, /*cpol*/0);
}
#endif

__device__ __forceinline__ void wait_async_lds() {
#if HAVE_ASYNC_LDS
#if __has_builtin(__builtin_amdgcn_s_wait_asynccnt)
    __builtin_amdgcn_s_wait_asynccnt(0);
#else
    asm volatile("s_wait_asynccnt 0" ::: "memory");
#endif
#endif
}

__global__ void lncc_zero_kernel(float* out) {
    if (threadIdx.x == 0 && blockIdx.x == 0) out[0] = 0.0f;
}

__global__ __launch_bounds__(NTHREADS)
void lncc_fused_kernel(const float* __restrict__ pred,
                       const float* __restrict__ targ,
                       float* __restrict__ out)
{
    __shared__ float sI[2][IN * IN];         // double-buffered pred plane + halo
    __shared__ float sJ[2][IN * IN];         // double-buffered target plane + halo
    __shared__ float sX[5][IN * TILE];       // x-filtered 5 fields (y 0..19, xo 0..15)
    __shared__ float sYr[5][5][TILE * TILE]; // z-ring of xy-filtered planes
    __shared__ float sRed[NTHREADS];

    const int tid  = threadIdx.x;
    const int lane = tid & 31;               // wave32
    const int wid  = tid >> 5;               // 8 waves
    const int x0 = blockIdx.x * TILE;
    const int y0 = blockIdx.y * TILE;
    const int bz = blockIdx.z;
    const int b  = bz >> 3;                  // batch (2)
    const int zb = (bz & 7) * ZCHUNK;        // z-chunk start

    const int planeStride = DIM * DIM;
    const int base = b * DIM * planeStride;  // C == 1

    // WMMA fragment coordinates (ISA 16x16 f32 layouts)
    const int m  = lane & 15;                // A row / D row base
    const int hi = (lane >> 4) & 1;          // lane half
    const int n  = lane & 15;                // B/D column

    float acc = 0.0f;

    // ---- plane-load issue (async global -> LDS, EXEC-masked OOB) -----------
    auto issue_plane = [&](int it) {
        const int z_in = zb - HALO + it;
        const bool zok = (z_in >= 0) && (z_in < DIM);
        float* bI = sI[it & 1];
        float* bJ = sJ[it & 1];
        const int zoff = base + z_in * planeStride;
        for (int p = tid; p < IN * IN; p += NTHREADS) {
            const int ly = p / IN, lx = p % IN;
            const int gy = y0 - HALO + ly;
            const int gx = x0 - HALO + lx;
            const bool inb = zok && (gy >= 0) && (gy < DIM) &&
                                    (gx >= 0) && (gx < DIM);
            const int gidx = zoff + gy * DIM + gx;
#if HAVE_ASYNC_LDS
            if (inb) {
                async_cp_g2l(pred + gidx, &bI[p]);
                async_cp_g2l(targ + gidx, &bJ[p]);
            } else {
                bI[p] = 0.0f;
                bJ[p] = 0.0f;
            }
#else
            bI[p] = inb ? pred[gidx] : 0.0f;
            bJ[p] = inb ? targ[gidx] : 0.0f;
#endif
        }
    };

    issue_plane(0);                          // prime the pipeline

    int slot = 0;                            // z-ring slot, avoids % 5
    for (int it = 0; it < NIT; ++it) {
        // ---- make plane `it` resident, then prefetch plane `it+1` ----------
        wait_async_lds();                    // only plane-`it` loads in flight
        __syncthreads();
        if (it + 1 < NIT)
            issue_plane(it + 1);             // overlaps with all compute below

        const float* bI = sI[it & 1];
        const float* bJ = sJ[it & 1];

        // ---- Phase 2: fused field compute + x-direction box sum ------------
        for (int p = tid; p < IN * TILE; p += NTHREADS) {
            const int ly = p / TILE, lx = p % TILE;
            const float* ri = &bI[ly * IN + lx];
            const float* rj = &bJ[ly * IN + lx];
            float si = 0.f, sj = 0.f, sii = 0.f, sjj = 0.f, sij = 0.f;
#pragma unroll
            for (int d = 0; d < KS; ++d) {
                const float iv = ri[d], jv = rj[d];
                si  += iv;      sj  += jv;
                sii += iv * iv; sjj += jv * jv; sij += iv * jv;
            }
            sX[0][p] = si;  sX[1][p] = sj;
            sX[2][p] = sii; sX[3][p] = sjj; sX[4][p] = sij;
        }
        __syncthreads();

        // ---- Phase 3: y-direction box sum as banded GEMM on matrix pipe ----
        // Out[16x16] = Band[16x20] x Xsum[20x16];  Band[yo][k] = (yo<=k<=yo+4)
        if (wid < 5) {                        // wave-uniform: EXEC all ones
            const float* Xf = sX[wid];
            float* dst = sYr[wid][slot];
#if HAVE_WMMA_F32
            v8f c = {0.f, 0.f, 0.f, 0.f, 0.f, 0.f, 0.f, 0.f};
#pragma unroll
            for (int kc = 0; kc < 5; ++kc) { // K = 20 in chunks of 4
                const int k0 = kc * 4 + (hi ? 2 : 0);
                const int k1 = kc * 4 + (hi ? 3 : 1);
                v2f a, bm;
                a.x  = (k0 >= m && k0 <= m + 4) ? 1.0f : 0.0f;
                a.y  = (k1 >= m && k1 <= m + 4) ? 1.0f : 0.0f;
                bm.x = Xf[k0 * TILE + n];
                bm.y = Xf[k1 * TILE + n];
                c = __builtin_amdgcn_wmma_f32_16x16x4_f32(
                        false, a, false, bm, (short)0, c, false, false);
            }
#pragma unroll
            for (int r = 0; r < 8; ++r)      // D: VGPR r -> row r (+8 for hi half)
                dst[(r + (hi ? 8 : 0)) * TILE + n] = c[r];
#else
            // VALU fallback (kept only so the file always compiles)
            for (int p = lane; p < TILE * TILE; p += 32) {
                const int yo = p / TILE, xo = p % TILE;
                float s = 0.f;
#pragma unroll
                for (int dy = 0; dy < KS; ++dy)
                    s += Xf[(yo + dy) * TILE + xo];
                dst[p] = s;
            }
#endif
        }
        __syncthreads();

        // ---- Phase 4: z-direction box sum (ring) + pointwise NCC -----------
        if (it >= 2 * HALO) {
            float s0 = 0.f, s1 = 0.f, s2 = 0.f, s3 = 0.f, s4 = 0.f;
#pragma unroll
            for (int s = 0; s < 5; ++s) {
                s0 += sYr[0][s][tid];
                s1 += sYr[1][s][tid];
                s2 += sYr[2][s][tid];
                s3 += sYr[3][s][tid];
                s4 += sYr[4][s][tid];
            }
            const float inv = 1.0f / 125.0f;
            const float uI = s0 * inv, uJ = s1 * inv;
            const float I2 = s2 * inv, J2 = s3 * inv, IJ = s4 * inv;
            const float cross = IJ - uI * uJ;
            const float vi = I2 - uI * uI;
            const float vj = J2 - uJ * uJ;
            acc += (cross * cross) / (vi * vj + 1e-5f);
        }

        slot = (slot == 4) ? 0 : slot + 1;
    }

    // ---- block reduction + scalar accumulate -------------------------------
    sRed[tid] = acc;
    __syncthreads();
#pragma unroll
    for (int off = NTHREADS / 2; off > 0; off >>= 1) {
        if (tid < off) sRed[tid] += sRed[tid + off];
        __syncthreads();
    }
    if (tid == 0) {
        const float scale = -1.0f / (2.0f * 192.0f * 192.0f * 192.0f);
        atomicAdd(out, sRed[0] * scale);
    }
}

extern "C" void kernel_launch(void* const* d_in, const int* in_sizes, int n_in,
                              void* d_out, int out_size, void* d_ws, size_t ws_size,
                              hipStream_t stream) {
    (void)in_sizes; (void)n_in; (void)out_size; (void)d_ws; (void)ws_size;
    const float* pred = (const float*)d_in[0];
    const float* targ = (const float*)d_in[1];
    float* out = (float*)d_out;

    lncc_zero_kernel<<<1, 1, 0, stream>>>(out);

    dim3 grid(DIM / TILE, DIM / TILE, 2 * (DIM / ZCHUNK));  // 12 x 12 x 16
    lncc_fused_kernel<<<grid, NTHREADS, 0, stream>>>(pred, targ, out);
}